// PeriodicTorsion_75419625717901
// MI455X (gfx1250) — compile-verified
//
#include <hip/hip_runtime.h>

typedef float v2f __attribute__((ext_vector_type(2)));
typedef float v8f __attribute__((ext_vector_type(8)));

#define MAIN_BLOCKS 4096
#define MAIN_THREADS 256

// ---------------------------------------------------------------------------
// Pass 1: per-torsion energy, block-level deterministic reduction -> partials
// ---------------------------------------------------------------------------
__global__ __launch_bounds__(MAIN_THREADS) void torsion_energy_kernel(
    const float* __restrict__ coords,       // [N_ATOMS,3] f32
    const long long* __restrict__ torsions, // [N_TORS,4] i64
    const float* __restrict__ fc,           // [N_TORS]
    const float* __restrict__ per,          // [N_TORS]
    const float* __restrict__ ph,           // [N_TORS]
    float* __restrict__ partial,            // [gridDim.x]
    int n) {
  const int tid = blockIdx.x * blockDim.x + threadIdx.x;
  const int stride = gridDim.x * blockDim.x;

  const longlong4* __restrict__ t4 = reinterpret_cast<const longlong4*>(torsions);

  float acc = 0.0f;

  for (int i = tid; i < n; i += stride) {
    // gfx1250: global_prefetch_b8 the next index row for this thread
    if (i + stride < n) {
      __builtin_prefetch((const void*)&t4[i + stride], 0, 1);
    }

    longlong4 t = t4[i];  // 2x global_load_b128

    const float* c0 = coords + 3 * (size_t)t.x;
    const float* c1 = coords + 3 * (size_t)t.y;
    const float* c2 = coords + 3 * (size_t)t.z;
    const float* c3 = coords + 3 * (size_t)t.w;

    float p0x = c0[0], p0y = c0[1], p0z = c0[2];
    float p1x = c1[0], p1y = c1[1], p1z = c1[2];
    float p2x = c2[0], p2y = c2[1], p2z = c2[2];
    float p3x = c3[0], p3y = c3[1], p3z = c3[2];

    float b1x = p1x - p0x, b1y = p1y - p0y, b1z = p1z - p0z;
    float b2x = p2x - p1x, b2y = p2y - p1y, b2z = p2z - p1z;
    float b3x = p3x - p2x, b3y = p3y - p2y, b3z = p3z - p2z;

    // n1 = b1 x b2 ; n2 = b2 x b3
    float n1x = b1y * b2z - b1z * b2y;
    float n1y = b1z * b2x - b1x * b2z;
    float n1z = b1x * b2y - b1y * b2x;
    float n2x = b2y * b3z - b2z * b3y;
    float n2y = b2z * b3x - b2x * b3z;
    float n2z = b2x * b3y - b2y * b3x;

    float dot12 = n1x * n2x + n1y * n2y + n1z * n2z;
    float nn1 = n1x * n1x + n1y * n1y + n1z * n1z;
    float nn2 = n2x * n2x + n2y * n2y + n2z * n2z;

    const float CLAMP = 0.999999999f;
    float cosv = dot12 / sqrtf(nn1 * nn2);
    cosv = fminf(fmaxf(cosv, -CLAMP), CLAMP);

    float dsg = n1x * b3x + n1y * b3y + n1z * b3z;
    float sgn = (dsg > 0.0f) ? 1.0f : ((dsg < 0.0f) ? -1.0f : 0.0f);

    float phi = acosf(cosv) * sgn;
    acc += fc[i] * (1.0f + cosf(per[i] * phi - ph[i]));
  }

  // wave32 shuffle reduction (warpSize == 32 on gfx1250)
  float v = acc;
  for (int off = 16; off > 0; off >>= 1) v += __shfl_down(v, off, 32);

  __shared__ float wsum[MAIN_THREADS / 32];
  const int lane = threadIdx.x & 31;
  const int wave = threadIdx.x >> 5;
  if (lane == 0) wsum[wave] = v;
  __syncthreads();

  if (threadIdx.x < (MAIN_THREADS / 32)) {
    float w = wsum[threadIdx.x];
    for (int off = (MAIN_THREADS / 64); off > 0; off >>= 1)
      w += __shfl_down(w, off, 32);
    if (threadIdx.x == 0) partial[blockIdx.x] = w;
  }
}

// ---------------------------------------------------------------------------
// Pass 2: one wave32 folds all partials into 64 slots (v2f A-matrix per lane),
// then a single V_WMMA_F32_16X16X4_F32 against an all-ones B collapses them.
// With B == 1, D[m][n] = rowsum(A,m) for every n, so lane0 (D[0..7][0]) plus
// lane16 (D[8..15][0]) gives the grand total. EXEC all-1s: 32 threads, no
// divergence before the WMMA.
// ---------------------------------------------------------------------------
__global__ __launch_bounds__(32) void final_reduce_wmma_kernel(
    const float* __restrict__ partial, int nparts, float* __restrict__ out) {
  const int lane = threadIdx.x;  // 0..31

  v2f a;
  a.x = 0.0f;
  a.y = 0.0f;
  for (int i = lane; i < nparts; i += 64) a.x += partial[i];
  for (int i = lane + 32; i < nparts; i += 64) a.y += partial[i];

  v2f b;
  b.x = 1.0f;
  b.y = 1.0f;
  v8f c = {};

  // (neg_a, A, neg_b, B, c_mod, C, reuse_a, reuse_b)
  v8f d = __builtin_amdgcn_wmma_f32_16x16x4_f32(
      false, a, false, b, (short)0, c, false, false);

  float s = d[0] + d[1] + d[2] + d[3] + d[4] + d[5] + d[6] + d[7];
  float tot = __shfl(s, 0, 32) + __shfl(s, 16, 32);
  if (lane == 0) out[0] = tot;
}

// ---------------------------------------------------------------------------
extern "C" void kernel_launch(void* const* d_in, const int* in_sizes, int n_in,
                              void* d_out, int out_size, void* d_ws,
                              size_t ws_size, hipStream_t stream) {
  const float* coords = (const float*)d_in[0];
  const long long* torsions = (const long long*)d_in[1];
  const float* fc = (const float*)d_in[2];
  const float* per = (const float*)d_in[3];
  const float* ph = (const float*)d_in[4];
  float* out = (float*)d_out;
  float* partial = (float*)d_ws;

  const int n = in_sizes[2];  // N_TORSIONS (fc is 1 element per torsion)

  int nblocks = MAIN_BLOCKS;
  size_t cap = ws_size / sizeof(float);
  if ((size_t)nblocks > cap) nblocks = (int)cap;
  if (nblocks < 1) nblocks = 1;

  torsion_energy_kernel<<<nblocks, MAIN_THREADS, 0, stream>>>(
      coords, torsions, fc, per, ph, partial, n);
  final_reduce_wmma_kernel<<<1, 32, 0, stream>>>(partial, nblocks, out);
}